// Distance_10960756539944
// MI455X (gfx1250) — compile-verified
//
#include <hip/hip_runtime.h>
#include <stdint.h>

#define ROWS 4096
#define COLS 8192
#define TC   16           // columns per block tile
#define NT   512          // threads per block (16 waves of 32)
#define NRG  64           // row groups (NT / (TC/2))
#define PT   64           // float2 elements per thread (64*2 = 128 floats)

typedef unsigned int v4u __attribute__((ext_vector_type(4)));
typedef int          v8i __attribute__((ext_vector_type(8)));
typedef int          v4i __attribute__((ext_vector_type(4)));

__global__ __launch_bounds__(NT, 1)
void sparsemax_cols_kernel(const float* __restrict__ x,
                           const float* __restrict__ a,
                           float* __restrict__ out)
{
    extern __shared__ float lds[];
    float* tile   = lds;                     // 65536 floats: [row][col], row-major, stride 16
    float* red_s  = lds + ROWS * TC;         // 1024 floats: per (rowgroup, col) partial sum / max
    float* red_c  = red_s + NRG * TC;        // 1024 floats: per (rowgroup, col) partial count
    float* colmax = red_c + NRG * TC;        // 16: per-column z max
    float* tauA   = colmax + TC;             // 16: per-column tau (shifted domain)
    float* cntA   = tauA + TC;               // 16: per-column previous support size
    float* chgA   = cntA + TC;               // 16: per-column "changed" flag

    const int tid  = (int)threadIdx.x;
    const int c0   = (int)blockIdx.x * TC;
    const int colp = (tid & 7) * 2;          // first of the two columns this thread owns
    const int rg   = tid >> 3;               // row group 0..63

    // ---- Tensor Data Mover: DMA the 16-col x 4096-row f32 tile into LDS ----
    // D# per CDNA5 ISA. 2D tile, X = 16 contiguous cols,
    // Y = 4096 rows with tensor_dim0_stride = 8192 elements. Issued by wave 0.
    if (tid < 32) {
        uint64_t ga = (uint64_t)(uintptr_t)(x + c0);
        uint32_t lo = (uint32_t)(uintptr_t)tile;     // low 32 bits of flat addr == LDS byte offset
        v4u g0;
        g0[0] = 1u;                                  // count=1 (valid user descriptor)
        g0[1] = lo;                                  // lds_addr
        g0[2] = (uint32_t)ga;                        // global_addr[95:64]
        g0[3] = ((uint32_t)(ga >> 32) & 0x01FFFFFFu) // global_addr[120:96]
                | (2u << 30);                        // type = 2 ("image")
        v8i g1;
        g1[0] = (2 << 16);                           // data_size = 2 -> 4 bytes
        g1[1] = (int)((unsigned)COLS << 16);         // tensor_dim0[15:0] in bits [63:48]
        g1[2] = (int)((unsigned)ROWS << 16);         // tensor_dim1[15:0] in bits [95:80]
        g1[3] = (TC << 16);                          // tile_dim0 in bits [127:112]
        g1[4] = ROWS;                                // tile_dim1 in bits [143:128]
        g1[5] = COLS;                                // tensor_dim0_stride[31:0]
        g1[6] = 0;                                   // stride hi / dim1_stride lo (unused, 2D)
        g1[7] = 0;
        v4i gz4 = {0, 0, 0, 0};                      // groups 2/3 unused (2D tensor)
        v8i gz8 = {0, 0, 0, 0, 0, 0, 0, 0};          // trailing group (unused)
        __builtin_amdgcn_tensor_load_to_lds(g0, g1, gz4, gz4, gz8, 0);
        __builtin_amdgcn_s_wait_tensorcnt(0);
    }
    __syncthreads();

    const float nscale = -expf(a[0]);                // z = nscale * x
    const float2* tile2 = (const float2*)tile;

    // ---- Pass 1: per-column max of z (conflict-free b64 LDS sweeps) ----
    float m0 = -3.402823466e38f, m1 = -3.402823466e38f;
    #pragma unroll 8
    for (int t = 0; t < PT; ++t) {
        float2 v = tile2[tid + NT * t];
        m0 = fmaxf(m0, nscale * v.x);
        m1 = fmaxf(m1, nscale * v.y);
    }
    red_s[rg * TC + colp]     = m0;
    red_s[rg * TC + colp + 1] = m1;
    __syncthreads();
    if (tid < TC) {
        float mm = red_s[tid];
        for (int g = 1; g < NRG; ++g) mm = fmaxf(mm, red_s[g * TC + tid]);
        colmax[tid] = mm;
        tauA[tid]   = -1.0f;   // shifted tau* is always in [-1, 0]
        cntA[tid]   = -1.0f;   // sentinel: never matches a real count
    }
    __syncthreads();

    // ---- Michelot fixed-point iteration (exact; ~3-5 passes for Gaussian data) ----
    for (int iter = 0; iter < 64; ++iter) {
        const float t0 = tauA[colp], t1 = tauA[colp + 1];
        const float z0m = colmax[colp], z1m = colmax[colp + 1];
        float s0 = 0.f, s1 = 0.f, n0 = 0.f, n1 = 0.f;
        #pragma unroll 8
        for (int t = 0; t < PT; ++t) {
            float2 v = tile2[tid + NT * t];
            float z0 = fmaf(nscale, v.x, -z0m);      // shifted z'
            float z1 = fmaf(nscale, v.y, -z1m);
            if (z0 > t0) { s0 += z0; n0 += 1.f; }
            if (z1 > t1) { s1 += z1; n1 += 1.f; }
        }
        red_s[rg * TC + colp]     = s0;
        red_s[rg * TC + colp + 1] = s1;
        red_c[rg * TC + colp]     = n0;
        red_c[rg * TC + colp + 1] = n1;
        __syncthreads();
        if (tid < TC) {
            float S = 0.f, C = 0.f;
            for (int g = 0; g < NRG; ++g) {
                S += red_s[g * TC + tid];
                C += red_c[g * TC + tid];
            }
            chgA[tid] = (C != cntA[tid]) ? 1.f : 0.f;  // support-size stability == converged
            cntA[tid] = C;
            tauA[tid] = (S - 1.0f) / C;                // C >= 1 always (max element active)
        }
        __syncthreads();
        float any = 0.f;
        for (int k = 0; k < TC; ++k) any += chgA[k];   // uniform across block
        if (any == 0.f) break;
        __syncthreads();
    }

    // ---- Output: max(z' - tau, 0), 64-bit coalesced stores ----
    const float t0 = tauA[colp], t1 = tauA[colp + 1];
    const float z0m = colmax[colp], z1m = colmax[colp + 1];
    float2* out2 = (float2*)out;
    const int ob = (c0 + colp) >> 1;
    #pragma unroll 8
    for (int t = 0; t < PT; ++t) {
        float2 v = tile2[tid + NT * t];
        float r0 = fmaxf(fmaf(nscale, v.x, -z0m) - t0, 0.f);
        float r1 = fmaxf(fmaf(nscale, v.y, -z1m) - t1, 0.f);
        int row = rg + NRG * t;
        out2[(size_t)row * (COLS / 2) + ob] = make_float2(r0, r1);
    }
}

extern "C" void kernel_launch(void* const* d_in, const int* in_sizes, int n_in,
                              void* d_out, int out_size, void* d_ws, size_t ws_size,
                              hipStream_t stream) {
    (void)in_sizes; (void)n_in; (void)out_size; (void)d_ws; (void)ws_size;
    const float* x = (const float*)d_in[0];
    const float* a = (const float*)d_in[1];   // scalar parameter (1 element)
    float* out = (float*)d_out;

    const size_t shmem = (size_t)(ROWS * TC + 2 * NRG * TC + 4 * TC) * sizeof(float); // ~264 KB

    // Allow >64KB dynamic LDS (WGP has 320KB on gfx1250). Harmless if it no-ops.
    static bool attr_done = false;
    if (!attr_done) {
        (void)hipFuncSetAttribute((const void*)sparsemax_cols_kernel,
                                  hipFuncAttributeMaxDynamicSharedMemorySize,
                                  (int)shmem);
        attr_done = true;
    }

    sparsemax_cols_kernel<<<dim3(COLS / TC), dim3(NT), shmem, stream>>>(x, a, out);
}